// HetSANN_43173011259536
// MI455X (gfx1250) — compile-verified
//
#include <hip/hip_runtime.h>
#include <hip/hip_bf16.h>
#include <math.h>

#define N_NODES 50000
#define N_EDGES 800000
#define T_TYPES 3
#define HEADS 8
#define SLOPE 0.2f

typedef __attribute__((ext_vector_type(2))) float v2f;
typedef __attribute__((ext_vector_type(8))) float v8f;

// ---- float atomic max via integer-representation ordering (handles mixed signs) ----
__device__ __forceinline__ void atomicMaxF(float* addr, float val) {
    if (val >= 0.0f) atomicMax((int*)addr, __float_as_int(val));
    else             atomicMin((unsigned int*)addr, (unsigned int)__float_as_int(val));
}

__global__ void fill_kernel(float* p, float v, int n) {
    int i = blockIdx.x * blockDim.x + threadIdx.x;
    if (i < n) p[i] = v;
}

// Bsum1[t][i][h] = sum_{d'} (al1[t][i][h*8+d'] + ar1[t][i][h*8+d'])
__global__ void prep_bsum1(const float* __restrict__ al1, const float* __restrict__ ar1,
                           float* __restrict__ Bsum1) {
    int i = blockIdx.x * blockDim.x + threadIdx.x;
    if (i >= T_TYPES * 64 * HEADS) return;
    int h = i % HEADS, row = (i / HEADS) % 64, t = i / (HEADS * 64);
    const float* al = al1 + ((size_t)t * 64 + row) * 64;
    const float* ar = ar1 + ((size_t)t * 64 + row) * 64;
    float s = 0.f;
    for (int d = 0; d < 8; ++d) s += al[h * 8 + d] + ar[h * 8 + d];
    Bsum1[((size_t)t * 64 + row) * HEADS + h] = s;
}

// c2[t][i] = sum_j (al2[t][i][j] + ar2[t][i][j])
__global__ void prep_c2(const float* __restrict__ al2, const float* __restrict__ ar2,
                        float* __restrict__ c2) {
    int i = blockIdx.x * blockDim.x + threadIdx.x;
    if (i >= T_TYPES * 16) return;
    int row = i % 16, t = i / 16;
    const float* al = al2 + ((size_t)t * 16 + row) * 16;
    const float* ar = ar2 + ((size_t)t * 16 + row) * 16;
    float s = 0.f;
    for (int j = 0; j < 16; ++j) s += al[j] + ar[j];
    c2[t * 16 + row] = s;
}

// Y[t] = X[n,64] @ W[t][64,64] via V_WMMA_F32_16X16X4_F32. block=128 (4 waves),
// wave w owns 16-col tile; grid.x covers 16-row tiles (n % 16 == 0), grid.z = type.
__global__ void gemm_x64_wmma(const float* __restrict__ X, const float* __restrict__ W,
                              float* __restrict__ Y, int n_rows) {
    int t = blockIdx.z;
    int wave = threadIdx.x >> 5, lane = threadIdx.x & 31;
    int half = lane >> 4, l = lane & 15;
    int row0 = blockIdx.x * 16, col0 = wave * 16;
    const float* Wt = W + (size_t)t * 64 * 64;
    const float* xr = X + (size_t)(row0 + l) * 64;
    v8f c = {0.f, 0.f, 0.f, 0.f, 0.f, 0.f, 0.f, 0.f};
#pragma unroll
    for (int kk = 0; kk < 16; ++kk) {
        int k = kk * 4 + half * 2;
        v2f a, b;
        a.x = xr[k];            a.y = xr[k + 1];              // A 16x4 frag
        b.x = Wt[k * 64 + col0 + l];
        b.y = Wt[(k + 1) * 64 + col0 + l];                    // B 4x16 frag
        c = __builtin_amdgcn_wmma_f32_16x16x4_f32(false, a, false, b, (short)0, c,
                                                  false, false);
    }
    float* Yt = Y + (size_t)t * n_rows * 64;
#pragma unroll
    for (int r = 0; r < 8; ++r)
        Yt[(size_t)(row0 + r + half * 8) * 64 + col0 + l] = c[r];
}

// Y[t] = X[n,64] @ W[t][64,16]. block=128: wave w -> row tile blockIdx.x*64 + w*16.
__global__ void gemm_x16_wmma(const float* __restrict__ X, const float* __restrict__ W,
                              float* __restrict__ Y, int n_rows) {
    int t = blockIdx.z;
    int wave = threadIdx.x >> 5, lane = threadIdx.x & 31;
    int half = lane >> 4, l = lane & 15;
    int row0 = blockIdx.x * 64 + wave * 16;
    if (row0 >= n_rows) return;                 // wave-uniform: EXEC stays all-1s
    const float* Wt = W + (size_t)t * 64 * 16;
    const float* xr = X + (size_t)(row0 + l) * 64;
    v8f c = {0.f, 0.f, 0.f, 0.f, 0.f, 0.f, 0.f, 0.f};
#pragma unroll
    for (int kk = 0; kk < 16; ++kk) {
        int k = kk * 4 + half * 2;
        v2f a, b;
        a.x = xr[k];           a.y = xr[k + 1];
        b.x = Wt[k * 16 + l];  b.y = Wt[(k + 1) * 16 + l];
        c = __builtin_amdgcn_wmma_f32_16x16x4_f32(false, a, false, b, (short)0, c,
                                                  false, false);
    }
    float* Yt = Y + (size_t)t * n_rows * 16;
#pragma unroll
    for (int r = 0; r < 8; ++r)
        Yt[(size_t)(row0 + r + half * 8) * 16 + l] = c[r];
}

// A1[t][n][h] = lrelu( Y1[t][n][:] . Bsum1[t][:][h] )
__global__ void scores1_kernel(const float* __restrict__ Y1, const float* __restrict__ Bsum1,
                               float* __restrict__ A1, int n) {
    __shared__ float Bs[64 * HEADS];
    int t = blockIdx.z;
    for (int i = threadIdx.x; i < 64 * HEADS; i += blockDim.x)
        Bs[i] = Bsum1[(size_t)t * 64 * HEADS + i];
    __syncthreads();
    int node = blockIdx.x * blockDim.x + threadIdx.x;
    if (node >= n) return;
    const float* y = Y1 + ((size_t)t * n + node) * 64;
    float acc[HEADS];
#pragma unroll
    for (int h = 0; h < HEADS; ++h) acc[h] = 0.f;
    for (int i = 0; i < 64; ++i) {
        float yv = y[i];
#pragma unroll
        for (int h = 0; h < HEADS; ++h) acc[h] += yv * Bs[i * HEADS + h];
    }
    float* out = A1 + ((size_t)t * n + node) * HEADS;
#pragma unroll
    for (int h = 0; h < HEADS; ++h) {
        float v = acc[h];
        out[h] = v > 0.f ? v : SLOPE * v;
    }
}

// A2[t][n] = lrelu( Y2[t][n][:] . c2[t][:] )
__global__ void scores2_kernel(const float* __restrict__ Y2, const float* __restrict__ c2,
                               float* __restrict__ A2, int n) {
    int t = blockIdx.z;
    int node = blockIdx.x * blockDim.x + threadIdx.x;
    if (node >= n) return;
    const float* y = Y2 + ((size_t)t * n + node) * 16;
    const float* cc = c2 + t * 16;
    float s = 0.f;
#pragma unroll
    for (int j = 0; j < 16; ++j) s += y[j] * cc[j];
    A2[(size_t)t * n + node] = s > 0.f ? s : SLOPE * s;
}

__global__ void segmax1_kernel(const int* __restrict__ src, const int* __restrict__ dst,
                               const int* __restrict__ et, const float* __restrict__ A1,
                               float* __restrict__ maxv, int e, int n) {
    int i = blockIdx.x * blockDim.x + threadIdx.x;
    if (i >= e) return;
    int s = src[i], d = dst[i], t = et[i];
    const float* a = A1 + ((size_t)t * n + s) * HEADS;
    float* m = maxv + (size_t)d * HEADS;
#pragma unroll
    for (int h = 0; h < HEADS; ++h) atomicMaxF(&m[h], a[h]);
}

__global__ void segsum1_kernel(const int* __restrict__ src, const int* __restrict__ dst,
                               const int* __restrict__ et, const float* __restrict__ A1,
                               const float* __restrict__ maxv, float* __restrict__ sumv,
                               int e, int n) {
    int i = blockIdx.x * blockDim.x + threadIdx.x;
    if (i >= e) return;
    int s = src[i], d = dst[i], t = et[i];
    const float* a = A1 + ((size_t)t * n + s) * HEADS;
    const float* m = maxv + (size_t)d * HEADS;
#pragma unroll
    for (int h = 0; h < HEADS; ++h)
        atomicAdd(&sumv[(size_t)d * HEADS + h], __expf(a[h] - m[h]));
}

// one thread per (edge, channel c in [0,64)); head = c>>3
__global__ void scatter1_kernel(const int* __restrict__ src, const int* __restrict__ dst,
                                const int* __restrict__ et, const float* __restrict__ A1,
                                const float* __restrict__ maxv, const float* __restrict__ sumv,
                                const float* __restrict__ Y1, float* __restrict__ out1,
                                int e, int n) {
    long idx = (long)blockIdx.x * blockDim.x + threadIdx.x;
    if (idx >= (long)e * 64) return;
    int c = (int)(idx & 63);
    int ed = (int)(idx >> 6);
    int s = __ldg(src + ed), d = __ldg(dst + ed), t = __ldg(et + ed);
    int h = c >> 3;
    float alpha = __expf(A1[((size_t)t * n + s) * HEADS + h] - maxv[(size_t)d * HEADS + h])
                  / sumv[(size_t)d * HEADS + h];
    atomicAdd(&out1[(size_t)d * 64 + c], alpha * Y1[((size_t)t * n + s) * 64 + c]);
}

__global__ void elu_kernel(float* __restrict__ p, long n) {
    long i = (long)blockIdx.x * blockDim.x + threadIdx.x;
    if (i < n) {
        float v = p[i];
        p[i] = v > 0.f ? v : expm1f(v);
    }
}

__global__ void segmax2_kernel(const int* __restrict__ src, const int* __restrict__ dst,
                               const int* __restrict__ et, const float* __restrict__ A2,
                               float* __restrict__ maxv, int e, int n) {
    int i = blockIdx.x * blockDim.x + threadIdx.x;
    if (i >= e) return;
    atomicMaxF(&maxv[dst[i]], A2[(size_t)et[i] * n + src[i]]);
}

__global__ void segsum2_kernel(const int* __restrict__ src, const int* __restrict__ dst,
                               const int* __restrict__ et, const float* __restrict__ A2,
                               const float* __restrict__ maxv, float* __restrict__ sumv,
                               int e, int n) {
    int i = blockIdx.x * blockDim.x + threadIdx.x;
    if (i >= e) return;
    int d = dst[i];
    atomicAdd(&sumv[d], __expf(A2[(size_t)et[i] * n + src[i]] - maxv[d]));
}

__global__ void scatter2_kernel(const int* __restrict__ src, const int* __restrict__ dst,
                                const int* __restrict__ et, const float* __restrict__ A2,
                                const float* __restrict__ maxv, const float* __restrict__ sumv,
                                const float* __restrict__ Y2, float* __restrict__ out2,
                                int e, int n) {
    long idx = (long)blockIdx.x * blockDim.x + threadIdx.x;
    if (idx >= (long)e * 16) return;
    int c = (int)(idx & 15);
    int ed = (int)(idx >> 4);
    int s = __ldg(src + ed), d = __ldg(dst + ed), t = __ldg(et + ed);
    float alpha = __expf(A2[(size_t)t * n + s] - maxv[d]) / sumv[d];
    atomicAdd(&out2[(size_t)d * 16 + c], alpha * Y2[((size_t)t * n + s) * 16 + c]);
}

// per-node residual + bias, block LDS reduction (ds_add_f32), then global accum
__global__ void final_kernel(const float* __restrict__ out2, const float* __restrict__ h0,
                             const float* __restrict__ res_w, const float* __restrict__ res_b,
                             float* __restrict__ accum, int n) {
    __shared__ float Ws[64 * 16];
    __shared__ float lacc[16];
    for (int i = threadIdx.x; i < 64 * 16; i += blockDim.x) Ws[i] = res_w[i];
    if (threadIdx.x < 16) lacc[threadIdx.x] = 0.f;
    __syncthreads();
    int node = blockIdx.x * blockDim.x + threadIdx.x;
    if (node < n) {
        const float* xv = h0 + (size_t)node * 64;
        float vals[16];
#pragma unroll
        for (int c = 0; c < 16; ++c) vals[c] = res_b[c] + out2[(size_t)node * 16 + c];
        for (int i = 0; i < 64; ++i) {
            float xi = xv[i];
#pragma unroll
            for (int c = 0; c < 16; ++c) vals[c] += xi * Ws[i * 16 + c];
        }
#pragma unroll
        for (int c = 0; c < 16; ++c) atomicAdd(&lacc[c], vals[c]);
    }
    __syncthreads();
    if (threadIdx.x < 16) atomicAdd(&accum[threadIdx.x], lacc[threadIdx.x]);
}

__global__ void write_out_kernel(const float* __restrict__ accum, float* __restrict__ out, int n) {
    int i = threadIdx.x;
    if (i < 16) out[i] = accum[i] / (float)n;
}

extern "C" void kernel_launch(void* const* d_in, const int* in_sizes, int n_in,
                              void* d_out, int out_size, void* d_ws, size_t ws_size,
                              hipStream_t stream) {
    (void)in_sizes; (void)n_in; (void)out_size; (void)ws_size;
    const float* x     = (const float*)d_in[0];
    const int*   src   = (const int*)d_in[1];
    const int*   dst   = (const int*)d_in[2];
    /* d_in[3] = ntype: unused by the reference */
    const int*   etype = (const int*)d_in[4];
    const float* al1   = (const float*)d_in[6];
    const float* ar1   = (const float*)d_in[7];
    const float* W1    = (const float*)d_in[5];
    const float* W2    = (const float*)d_in[8];
    const float* al2   = (const float*)d_in[9];
    const float* ar2   = (const float*)d_in[10];
    const float* res_w = (const float*)d_in[11];
    const float* res_b = (const float*)d_in[12];
    float* out = (float*)d_out;

    const int n = N_NODES, e = N_EDGES;

    float* ws = (float*)d_ws;
    size_t off = 0;
    auto alloc = [&](size_t cnt) {
        float* p = ws + off;
        off += (cnt + 63) & ~(size_t)63;
        return p;
    };
    float* Y1    = alloc((size_t)T_TYPES * n * 64);
    float* A1    = alloc((size_t)T_TYPES * n * HEADS);
    float* maxv1 = alloc((size_t)n * HEADS);
    float* sumv1 = alloc((size_t)n * HEADS);
    float* h0    = alloc((size_t)n * 64);      // scatter accum, then ELU in place
    float* Y2    = alloc((size_t)T_TYPES * n * 16);
    float* A2    = alloc((size_t)T_TYPES * n);
    float* maxv2 = alloc((size_t)n);
    float* sumv2 = alloc((size_t)n);
    float* out2  = alloc((size_t)n * 16);
    float* Bsum1 = alloc((size_t)T_TYPES * 64 * HEADS);
    float* c2    = alloc((size_t)T_TYPES * 16);
    float* accum = alloc((size_t)16);

    // ---- init scratch (memset nodes are graph-capturable) ----
    hipMemsetAsync(sumv1, 0, (size_t)n * HEADS * sizeof(float), stream);
    hipMemsetAsync(h0,    0, (size_t)n * 64 * sizeof(float), stream);
    hipMemsetAsync(sumv2, 0, (size_t)n * sizeof(float), stream);
    hipMemsetAsync(out2,  0, (size_t)n * 16 * sizeof(float), stream);
    hipMemsetAsync(accum, 0, 16 * sizeof(float), stream);
    fill_kernel<<<(n * HEADS + 255) / 256, 256, 0, stream>>>(maxv1, -INFINITY, n * HEADS);
    fill_kernel<<<(n + 255) / 256, 256, 0, stream>>>(maxv2, -INFINITY, n);
    prep_bsum1<<<(T_TYPES * 64 * HEADS + 255) / 256, 256, 0, stream>>>(al1, ar1, Bsum1);
    prep_c2<<<1, 64, 0, stream>>>(al2, ar2, c2);

    // ---- layer 0 (8 heads, ELU) ----
    gemm_x64_wmma<<<dim3(n / 16, 1, T_TYPES), 128, 0, stream>>>(x, W1, Y1, n);
    scores1_kernel<<<dim3((n + 255) / 256, 1, T_TYPES), 256, 0, stream>>>(Y1, Bsum1, A1, n);
    segmax1_kernel<<<(e + 255) / 256, 256, 0, stream>>>(src, dst, etype, A1, maxv1, e, n);
    segsum1_kernel<<<(e + 255) / 256, 256, 0, stream>>>(src, dst, etype, A1, maxv1, sumv1, e, n);
    scatter1_kernel<<<(int)(((long)e * 64 + 255) / 256), 256, 0, stream>>>(
        src, dst, etype, A1, maxv1, sumv1, Y1, h0, e, n);
    elu_kernel<<<(int)(((long)n * 64 + 255) / 256), 256, 0, stream>>>(h0, (long)n * 64);

    // ---- layer 1 (1 head, residual, no act) ----
    gemm_x16_wmma<<<dim3((n / 16 + 3) / 4, 1, T_TYPES), 128, 0, stream>>>(h0, W2, Y2, n);
    scores2_kernel<<<dim3((n + 255) / 256, 1, T_TYPES), 256, 0, stream>>>(Y2, c2, A2, n);
    segmax2_kernel<<<(e + 255) / 256, 256, 0, stream>>>(src, dst, etype, A2, maxv2, e, n);
    segsum2_kernel<<<(e + 255) / 256, 256, 0, stream>>>(src, dst, etype, A2, maxv2, sumv2, e, n);
    scatter2_kernel<<<(int)(((long)e * 16 + 255) / 256), 256, 0, stream>>>(
        src, dst, etype, A2, maxv2, sumv2, Y2, out2, e, n);
    final_kernel<<<(n + 255) / 256, 256, 0, stream>>>(out2, h0, res_w, res_b, accum, n);
    write_out_kernel<<<1, 16, 0, stream>>>(accum, out, n);
}